// CellLineKGModel_47081431498871
// MI455X (gfx1250) — compile-verified
//
#include <hip/hip_runtime.h>

// ---------------------------------------------------------------------------
// CDNA5 (gfx1250) CellLineKG forward pass.
//  - Dense linears: V_WMMA_F32_16X16X4_F32 (fp32-exact matrix path),
//    B (weights) staged per-block into LDS via GLOBAL_LOAD_ASYNC_TO_LDS_B128
//    (ASYNCcnt) so all 8 waves of a block share one copy of the weight chunk.
//  - Edge segment-sums: coalesced float4 gathers + hardware fp32 atomics
//    (L2-resident working set: hp/nb are 10 MB each vs 192 MB L2).
// ---------------------------------------------------------------------------

typedef __attribute__((ext_vector_type(2))) float v2f;
typedef __attribute__((ext_vector_type(8))) float v8f;

static constexpr int HD    = 128;    // hidden dim
static constexpr int NPROT = 20000;
static constexpr int KC    = 64;     // K-chunk staged in LDS (64*128*4 = 32 KB)

#define USE_ASYNC_LDS 1

__device__ __forceinline__ v8f wmma4(v2f a, v2f b, v8f c) {
  // D = A(16x4,f32) * B(4x16,f32) + C(16x16,f32)
  return __builtin_amdgcn_wmma_f32_16x16x4_f32(
      /*neg_a=*/false, a, /*neg_b=*/false, b,
      /*c_mod=*/(short)0, c, /*reuse_a=*/false, /*reuse_b=*/false);
}

__device__ __forceinline__ void atomic_add_f32(float* p, float v) {
  __hip_atomic_fetch_add(p, v, __ATOMIC_RELAXED, __HIP_MEMORY_SCOPE_AGENT);
}

// Cooperative copy of nfloats (multiple of 1024) from global to LDS.
// Async path: per-lane GLOBAL_LOAD_ASYNC_TO_LDS_B128, ASYNCcnt-tracked;
// LDS dest address = low 32 bits of the flat shared pointer (ISA 10.2).
__device__ __forceinline__ void stage_lds(const float* __restrict__ gsrc,
                                          float* __restrict__ lds, int nfloats) {
  const int n4 = nfloats >> 2;
#if USE_ASYNC_LDS
  for (int i = threadIdx.x; i < n4; i += blockDim.x) {
    const unsigned int  laddr = (unsigned int)(unsigned long long)(lds + i * 4);
    const unsigned long long gaddr = (unsigned long long)(gsrc + i * 4);
    asm volatile("global_load_async_to_lds_b128 %0, %1, off"
                 :: "v"(laddr), "v"(gaddr) : "memory");
  }
  asm volatile("s_wait_asynccnt 0x0" ::: "memory");
#else
  const float4* s = (const float4*)gsrc;
  float4* d = (float4*)lds;
  for (int i = threadIdx.x; i < n4; i += blockDim.x) d[i] = s[i];
#endif
  __syncthreads();
}

// ---------------------------------------------------------------------------
// out[M,128] = X[M,D] @ W[D,128] + bias ; optional duplicate write to out2.
// Block = 8 waves sharing the LDS-staged W chunk; one wave per 16-row stripe.
// A frag (16x4 f32): lane<16 -> K+0,K+1 ; lane>=16 -> K+2,K+3 ; row = lane&15.
// C/D: vgpr r -> row r (+8 for lanes>=16), col = lane&15.
// NOTE: no early returns (block-wide barriers); partial tiles use clamped
// loads + masked stores, keeping EXEC all-ones for WMMA.
// ---------------------------------------------------------------------------
__global__ void embed_gemm(const float* __restrict__ X, const float* __restrict__ W,
                           const float* __restrict__ bias,
                           float* __restrict__ out, float* __restrict__ out2,
                           int M, int D) {
  __shared__ float Bsm[KC * HD];

  const int lane  = threadIdx.x & 31;
  const int wave  = threadIdx.x >> 5;
  const int mtile = blockIdx.x * (blockDim.x >> 5) + wave;
  const int mbase = mtile * 16;

  const int row  = lane & 15;
  const int koff = (lane >> 4) * 2;
  const int arow = min(mbase + row, M - 1);   // clamp for partial/invalid tiles

  v8f acc[8];
#pragma unroll
  for (int t = 0; t < 8; ++t) {
    const float bv = bias[t * 16 + row];      // bias is per-column
#pragma unroll
    for (int r = 0; r < 8; ++r) acc[t][r] = bv;
  }

  const float* Arow = X + (size_t)arow * D;
  for (int k0 = 0; k0 < D; k0 += KC) {
    const int kc = min(KC, D - k0);           // always a multiple of 16
    stage_lds(W + (size_t)k0 * HD, Bsm, kc * HD);
    for (int k = 0; k < kc; k += 4) {
      const v2f a = *(const v2f*)(Arow + k0 + k + koff);
      const float* B0 = Bsm + (k + koff) * HD;
#pragma unroll
      for (int t = 0; t < 8; ++t) {
        v2f b;
        b.x = B0[t * 16 + row];
        b.y = B0[HD + t * 16 + row];
        acc[t] = wmma4(a, b, acc[t]);
      }
    }
    __syncthreads();
  }

  const int mr = mbase + ((lane >> 4) << 3);
#pragma unroll
  for (int r = 0; r < 8; ++r) {
    const int m = mr + r;
    if (m < M) {
#pragma unroll
      for (int t = 0; t < 8; ++t) {
        const float v = acc[t][r];
        out[(size_t)m * HD + t * 16 + row] = v;
        if (out2) out2[(size_t)m * HD + t * 16 + row] = v;
      }
    }
  }
}

// ---------------------------------------------------------------------------
// SAGE layer: hp[m,:] += relu(hp@Wself + bself + (neigh*invdeg)@Wneigh)[m,:]
// In-place safe: each wave reads/writes only its own 16 rows, all reads of
// those rows complete before the epilogue writes.
// ---------------------------------------------------------------------------
__global__ void sage_layer(float* __restrict__ hp, const float* __restrict__ neigh,
                           const float* __restrict__ invdeg,
                           const float* __restrict__ Wself,
                           const float* __restrict__ bself,
                           const float* __restrict__ Wneigh) {
  __shared__ float Bsm[KC * HD];

  const int lane  = threadIdx.x & 31;
  const int wave  = threadIdx.x >> 5;
  const int mtile = blockIdx.x * (blockDim.x >> 5) + wave;
  const int mbase = mtile * 16;

  const int row  = lane & 15;
  const int koff = (lane >> 4) * 2;
  const int arow = min(mbase + row, NPROT - 1);

  v8f acc[8];
#pragma unroll
  for (int t = 0; t < 8; ++t) {
    const float bv = bself[t * 16 + row];
#pragma unroll
    for (int r = 0; r < 8; ++r) acc[t][r] = bv;
  }

  // self term: hp @ Wself
  const float* As = hp + (size_t)arow * HD;
  for (int k0 = 0; k0 < HD; k0 += KC) {
    stage_lds(Wself + (size_t)k0 * HD, Bsm, KC * HD);
    for (int k = 0; k < KC; k += 4) {
      const v2f a = *(const v2f*)(As + k0 + k + koff);
      const float* B0 = Bsm + (k + koff) * HD;
#pragma unroll
      for (int t = 0; t < 8; ++t) {
        v2f b;
        b.x = B0[t * 16 + row];
        b.y = B0[HD + t * 16 + row];
        acc[t] = wmma4(a, b, acc[t]);
      }
    }
    __syncthreads();
  }

  // mean-aggregated neighbor term: (neigh * invdeg) @ Wneigh  (scale per A-row)
  const float s = invdeg[arow];
  const float* An = neigh + (size_t)arow * HD;
  for (int k0 = 0; k0 < HD; k0 += KC) {
    stage_lds(Wneigh + (size_t)k0 * HD, Bsm, KC * HD);
    for (int k = 0; k < KC; k += 4) {
      v2f a = *(const v2f*)(An + k0 + k + koff);
      a.x *= s; a.y *= s;
      const float* B0 = Bsm + (k + koff) * HD;
#pragma unroll
      for (int t = 0; t < 8; ++t) {
        v2f b;
        b.x = B0[t * 16 + row];
        b.y = B0[HD + t * 16 + row];
        acc[t] = wmma4(a, b, acc[t]);
      }
    }
    __syncthreads();
  }

  // ReLU + residual (in place)
  const int mr = mbase + ((lane >> 4) << 3);
#pragma unroll
  for (int r = 0; r < 8; ++r) {
    const int m = mr + r;
    if (m < NPROT) {
#pragma unroll
      for (int t = 0; t < 8; ++t) {
        const float v = fmaxf(acc[t][r], 0.0f);
        hp[(size_t)m * HD + t * 16 + row] += v;
      }
    }
  }
}

// ---------------------------------------------------------------------------
// GCN (norm='both') layer: hp[m,:] += relu((agg@W)*ddinv[m] + b)
// ---------------------------------------------------------------------------
__global__ void gcn_layer(float* __restrict__ hp, const float* __restrict__ agg,
                          const float* __restrict__ ddinv,
                          const float* __restrict__ W,
                          const float* __restrict__ bias) {
  __shared__ float Bsm[KC * HD];

  const int lane  = threadIdx.x & 31;
  const int wave  = threadIdx.x >> 5;
  const int mtile = blockIdx.x * (blockDim.x >> 5) + wave;
  const int mbase = mtile * 16;

  const int row  = lane & 15;
  const int koff = (lane >> 4) * 2;
  const int arow = min(mbase + row, NPROT - 1);

  v8f acc[8];
#pragma unroll
  for (int t = 0; t < 8; ++t)
#pragma unroll
    for (int r = 0; r < 8; ++r) acc[t][r] = 0.0f;

  const float* Aa = agg + (size_t)arow * HD;
  for (int k0 = 0; k0 < HD; k0 += KC) {
    stage_lds(W + (size_t)k0 * HD, Bsm, KC * HD);
    for (int k = 0; k < KC; k += 4) {
      const v2f a = *(const v2f*)(Aa + k0 + k + koff);
      const float* B0 = Bsm + (k + koff) * HD;
#pragma unroll
      for (int t = 0; t < 8; ++t) {
        v2f b;
        b.x = B0[t * 16 + row];
        b.y = B0[HD + t * 16 + row];
        acc[t] = wmma4(a, b, acc[t]);
      }
    }
    __syncthreads();
  }

  const int mr = mbase + ((lane >> 4) << 3);
#pragma unroll
  for (int r = 0; r < 8; ++r) {
    const int m = mr + r;
    if (m < NPROT) {
      const float dsc = ddinv[m];
#pragma unroll
      for (int t = 0; t < 8; ++t) {
        const float v = fmaxf(acc[t][r] * dsc + bias[t * 16 + row], 0.0f);
        hp[(size_t)m * HD + t * 16 + row] += v;
      }
    }
  }
}

// ---------------------------------------------------------------------------
// Sparse helpers
// ---------------------------------------------------------------------------
__global__ void zero_f32(float* __restrict__ p, int n) {
  const int i = blockIdx.x * blockDim.x + threadIdx.x;
  if (i < n) p[i] = 0.0f;
}

__global__ void deg_count(const int* __restrict__ idx, float* __restrict__ deg, int E) {
  const int i = blockIdx.x * blockDim.x + threadIdx.x;
  if (i < E) atomic_add_f32(&deg[idx[i]], 1.0f);
}

// mode 0: deg>0 ? 1/deg : 0   (SAGE mean)
// mode 1: rsqrt(max(deg,1))   (GCN norm='both')
__global__ void deg_transform(float* __restrict__ deg, int n, int mode) {
  const int i = blockIdx.x * blockDim.x + threadIdx.x;
  if (i >= n) return;
  const float d = deg[i];
  deg[i] = (mode == 0) ? (d > 0.0f ? 1.0f / d : 0.0f) : rsqrtf(fmaxf(d, 1.0f));
}

// One wave per edge; lane moves float4 (128 floats / 32 lanes * 4).
// Optional per-source scale folds deg_src^-0.5 into the CLP gather.
__global__ void edge_scatter(const int* __restrict__ src, const int* __restrict__ dst,
                             const float* __restrict__ feat,
                             const float* __restrict__ scale,
                             float* __restrict__ outbuf, int E) {
  const int t = blockIdx.x * blockDim.x + threadIdx.x;
  const int e = t >> 5;
  if (e >= E) return;
  const int lane = t & 31;
  const int s = src[e];
  const int d = dst[e];
  const float sc = scale ? scale[s] : 1.0f;
  const float4 v = *(const float4*)(feat + (size_t)s * HD + lane * 4);
  float* o = outbuf + (size_t)d * HD + lane * 4;
  atomic_add_f32(o + 0, v.x * sc);
  atomic_add_f32(o + 1, v.y * sc);
  atomic_add_f32(o + 2, v.z * sc);
  atomic_add_f32(o + 3, v.w * sc);
}

__global__ void add_inplace(float* __restrict__ out, const float* __restrict__ a, int n) {
  const int i = blockIdx.x * blockDim.x + threadIdx.x;
  if (i < n) out[i] += a[i];
}

// ---------------------------------------------------------------------------
// Host orchestration
// ---------------------------------------------------------------------------
static inline int cdiv(int a, int b) { return (a + b - 1) / b; }

extern "C" void kernel_launch(void* const* d_in, const int* in_sizes, int n_in,
                              void* d_out, int out_size, void* d_ws, size_t ws_size,
                              hipStream_t stream) {
  (void)n_in; (void)out_size; (void)ws_size;

  const float* drugF = (const float*)d_in[0];
  const float* protF = (const float*)d_in[1];
  const float* cellF = (const float*)d_in[2];
  const float* disF  = (const float*)d_in[3];
  const int* pp_src  = (const int*)d_in[4];
  const int* pp_dst  = (const int*)d_in[5];
  const int* clp_src = (const int*)d_in[6];
  const int* clp_dst = (const int*)d_in[7];
  const float* Wd   = (const float*)d_in[8];
  const float* bd   = (const float*)d_in[9];
  const float* Wp   = (const float*)d_in[10];
  const float* bp   = (const float*)d_in[11];
  const float* Wc   = (const float*)d_in[12];
  const float* bc   = (const float*)d_in[13];
  const float* Wdis = (const float*)d_in[14];
  const float* bdis = (const float*)d_in[15];
  const float* sage_Wself  = (const float*)d_in[16];
  const float* sage_bself  = (const float*)d_in[17];
  const float* sage_Wneigh = (const float*)d_in[18];
  const float* gcn_W = (const float*)d_in[19];
  const float* gcn_b = (const float*)d_in[20];

  const int N_DRUG = 2000, N_CELL = 1000, N_DIS = 5000;
  const int D_DRUG = 1024, D_PROT = 1280, D_CELL = 2000, D_DIS = 512;
  const int E_PP  = in_sizes[4];
  const int E_CLP = in_sizes[6];

  float* out      = (float*)d_out;
  float* out_drug = out;
  float* out_prot = out_drug + (size_t)N_DRUG * HD;
  float* out_cell = out_prot + (size_t)NPROT * HD;
  float* out_dis  = out_cell + (size_t)N_CELL * HD;

  char* ws = (char*)d_ws;
  float* hp    = (float*)ws;                                   // [NPROT,HD]
  float* nb    = (float*)(ws + (size_t)NPROT * HD * 4);        // [NPROT,HD] neigh/agg
  float* degpp = (float*)(ws + 2 * (size_t)NPROT * HD * 4);    // [NPROT]
  float* dsrc  = degpp + NPROT;                                // [N_CELL]
  float* ddst  = dsrc + N_CELL;                                // [NPROT]

  const int TB = 256;                 // 8 waves/block
  const int WPB = TB / 32;

  // ---- embeddings (WMMA GEMMs). Protein also copied to out_prot as h_init.
  embed_gemm<<<cdiv(cdiv(N_DRUG, 16), WPB), TB, 0, stream>>>(drugF, Wd, bd, out_drug, nullptr, N_DRUG, D_DRUG);
  embed_gemm<<<cdiv(cdiv(NPROT, 16), WPB), TB, 0, stream>>>(protF, Wp, bp, hp, out_prot, NPROT, D_PROT);
  embed_gemm<<<cdiv(cdiv(N_CELL, 16), WPB), TB, 0, stream>>>(cellF, Wc, bc, out_cell, nullptr, N_CELL, D_CELL);
  embed_gemm<<<cdiv(cdiv(N_DIS, 16), WPB), TB, 0, stream>>>(disF, Wdis, bdis, out_dis, nullptr, N_DIS, D_DIS);

  // ---- protein-protein mean-aggregation degree
  zero_f32<<<cdiv(NPROT, TB), TB, 0, stream>>>(degpp, NPROT);
  deg_count<<<cdiv(E_PP, TB), TB, 0, stream>>>(pp_dst, degpp, E_PP);
  deg_transform<<<cdiv(NPROT, TB), TB, 0, stream>>>(degpp, NPROT, 0);

  // ---- 2 SAGE layers
  for (int l = 0; l < 2; ++l) {
    zero_f32<<<cdiv(NPROT * HD, TB), TB, 0, stream>>>(nb, NPROT * HD);
    edge_scatter<<<cdiv(E_PP * 32, TB), TB, 0, stream>>>(pp_src, pp_dst, hp, nullptr, nb, E_PP);
    sage_layer<<<cdiv(cdiv(NPROT, 16), WPB), TB, 0, stream>>>(
        hp, nb, degpp, sage_Wself + (size_t)l * HD * HD,
        sage_bself + (size_t)l * HD, sage_Wneigh + (size_t)l * HD * HD);
  }

  // ---- bipartite GraphConv norm='both'
  zero_f32<<<cdiv(N_CELL, TB), TB, 0, stream>>>(dsrc, N_CELL);
  zero_f32<<<cdiv(NPROT, TB), TB, 0, stream>>>(ddst, NPROT);
  deg_count<<<cdiv(E_CLP, TB), TB, 0, stream>>>(clp_src, dsrc, E_CLP);
  deg_count<<<cdiv(E_CLP, TB), TB, 0, stream>>>(clp_dst, ddst, E_CLP);
  deg_transform<<<cdiv(N_CELL, TB), TB, 0, stream>>>(dsrc, N_CELL, 1);
  deg_transform<<<cdiv(NPROT, TB), TB, 0, stream>>>(ddst, NPROT, 1);

  zero_f32<<<cdiv(NPROT * HD, TB), TB, 0, stream>>>(nb, NPROT * HD);
  edge_scatter<<<cdiv(E_CLP * 32, TB), TB, 0, stream>>>(clp_src, clp_dst, out_cell, dsrc, nb, E_CLP);
  gcn_layer<<<cdiv(cdiv(NPROT, 16), WPB), TB, 0, stream>>>(hp, nb, ddst, gcn_W, gcn_b);

  // ---- final residual: out_prot currently holds h_init; hp = full updated state
  add_inplace<<<cdiv(NPROT * HD, TB), TB, 0, stream>>>(out_prot, hp, NPROT * HD);
}